// Graph_Directed_A_29978871726196
// MI455X (gfx1250) — compile-verified
//
#include <hip/hip_runtime.h>
#include <math.h>
#include <stdint.h>

#define NODES 1024
#define NN (NODES * NODES)
#define IN_DIM 128

typedef __attribute__((ext_vector_type(2))) float v2f;
typedef __attribute__((ext_vector_type(8))) float v8f;

// ---------------------------------------------------------------- utilities
__global__ void zero_f32(float* __restrict__ p, int n) {
    int i = blockIdx.x * blockDim.x + threadIdx.x;
    int stride = gridDim.x * blockDim.x;
    for (; i < n; i += stride) p[i] = 0.0f;
}

// p[n] = dot(features[n,:], w) + b      (one 128-thread block per node)
__global__ void node_potential(const float* __restrict__ feat,
                               const float* __restrict__ w,
                               const float* __restrict__ b,
                               float* __restrict__ p) {
    __shared__ float red[IN_DIM];
    const int n = blockIdx.x, t = threadIdx.x;
    red[t] = feat[n * IN_DIM + t] * w[t];
    __syncthreads();
    for (int s = IN_DIM / 2; s > 0; s >>= 1) {
        if (t < s) red[t] += red[t + s];
        __syncthreads();
    }
    if (t == 0) p[n] = red[0] + b[0];
}

// Scatter edges into dense A (val>0 -> (u,v),(v,u) w=1+val ; else self loops w=1)
__global__ void edge_scatter(const int* __restrict__ eu, const int* __restrict__ ev,
                             const float* __restrict__ p, float* __restrict__ A, int M) {
    int e = blockIdx.x * blockDim.x + threadIdx.x;
    if (e >= M) return;
    const int u = eu[e], v = ev[e];
    const float val = p[u] - p[v];
    if (val > 0.0f) {
        const float w = 1.0f + val;
        atomicAdd(&A[u * NODES + v], w);
        atomicAdd(&A[v * NODES + u], w);
    } else {
        atomicAdd(&A[u * NODES + u], 1.0f);
        atomicAdd(&A[v * NODES + v], 1.0f);
    }
}

__global__ void add_eye(float* __restrict__ A) {
    int i = blockIdx.x * blockDim.x + threadIdx.x;
    if (i < NODES) A[i * NODES + i] += 1.0f;
}

// dinv[j] = (sum_i A_sym[i,j])^{-1/2},  A_sym = 0.5(A + A^T)
__global__ void degree_inv_sqrt(const float* __restrict__ A, float* __restrict__ dinv) {
    __shared__ float red[256];
    const int j = blockIdx.x, t = threadIdx.x;
    float s = 0.0f;
    for (int i = t; i < NODES; i += 256)
        s += A[j * NODES + i] + A[i * NODES + j];
    red[t] = s;
    __syncthreads();
    for (int k = 128; k > 0; k >>= 1) {
        if (t < k) red[t] += red[t + k];
        __syncthreads();
    }
    if (t == 0) {
        float deg = 0.5f * red[0];
        if (deg == 0.0f) deg = 1.0f;
        dinv[j] = 1.0f / sqrtf(deg);
    }
}

// L = -exp(i*(pi/2)*(A - A^T)) .* (d_i d_j 0.5(A+A^T)) ; write T0=I, T1=L
__global__ void build_L(const float* __restrict__ A, const float* __restrict__ dinv,
                        float* __restrict__ out) {
    const int idx = blockIdx.x * blockDim.x + threadIdx.x;
    const int i = idx >> 10, j = idx & (NODES - 1);
    const float a = A[idx];
    const float at = A[j * NODES + i];
    const float asym = 0.5f * (a + at) * dinv[i] * dinv[j];
    const float ph = 1.57079632679489662f * (a - at);
    const float sn = sinf(ph), cs = cosf(ph);
    out[0 * (size_t)NN + idx] = (i == j) ? 1.0f : 0.0f; // T0 real
    out[5 * (size_t)NN + idx] = 0.0f;                   // T0 imag
    out[1 * (size_t)NN + idx] = -cs * asym;             // T1 real = L real
    out[6 * (size_t)NN + idx] = -sn * asym;             // T1 imag = L imag
}

// ------------------------------------------------- complex Chebyshev GEMM
// D = 2 * (L @ P) - Q     (complex, f32, via V_WMMA_F32_16X16X4_F32)
// Tiles staged with async global->LDS copies (ASYNCcnt), double-buffered.
#define BM 64
#define BN 64
#define BK 16
#define AP 20   // LDS pitch for 64x16 A tiles (floats, float4-aligned, 16 banks)
#define BP 80   // LDS pitch for 16x64 B tiles (floats)

__device__ __forceinline__ v8f wmma4(v2f a, v2f b, v8f c) {
    // f32 WMMA: NEG/NEG_HI must be zero for A/B (only C-neg/abs exist)
    return __builtin_amdgcn_wmma_f32_16x16x4_f32(false, a, false, b, (short)0, c,
                                                 false, false);
}

// async DMA: 16 bytes global -> LDS, tracked by ASYNCcnt
__device__ __forceinline__ void async_b128(const float* lds, const float* g) {
    uint32_t loff = (uint32_t)(uintptr_t)lds;      // low 32 bits = LDS offset
    uint64_t gaddr = (uint64_t)(uintptr_t)g;
    asm volatile("global_load_async_to_lds_b128 %0, %1, off"
                 :: "v"(loff), "v"(gaddr) : "memory");
}
__device__ __forceinline__ void wait_async_le4() {
    asm volatile("s_wait_asynccnt 4" ::: "memory");
}
__device__ __forceinline__ void wait_async_0() {
    asm volatile("s_wait_asynccnt 0" ::: "memory");
}

__global__ __launch_bounds__(256)
void cheb_cgemm(const float* __restrict__ Lre, const float* __restrict__ Lim,
                const float* __restrict__ Pre, const float* __restrict__ Pim,
                const float* __restrict__ Qre, const float* __restrict__ Qim,
                float* __restrict__ Dre, float* __restrict__ Dim) {
    __shared__ float sA[2][2][BM * AP]; // [buf][re/im][64x16 tile]
    __shared__ float sB[2][2][BK * BP]; // [buf][re/im][16x64 tile]

    const int tid = threadIdx.x;
    const int wave = tid >> 5;
    const int lane = tid & 31;
    const int half = lane >> 4;  // K-half for A/B fragments, M-half for C/D
    const int lh = lane & 15;    // M row (A), N col (B/C/D)
    const int bm = blockIdx.x * BM;
    const int bn = blockIdx.y * BN;
    const int wcol = (wave & 3) * 16;  // wave's N offset in block tile
    const int wrow = (wave >> 2) * 32; // wave's M offset (two 16-row subtiles)

    // per-thread staging coordinates (one float4 per tile per thread)
    const int ar = tid >> 2, ac = (tid & 3) * 4;   // A: 64 rows x 16 cols
    const int br = tid >> 4, bc = (tid & 15) * 4;  // B: 16 rows x 64 cols

    v8f acc[4] = {}; // {re0, im0, re1, im1}

    const int NCH = NODES / BK; // 64 K-chunks

    // prologue: stage chunk 0 into buffer 0
    {
        const int g = (bm + ar) * NODES + ac;
        async_b128(&sA[0][0][ar * AP + ac], Lre + g);
        async_b128(&sA[0][1][ar * AP + ac], Lim + g);
        const int h = br * NODES + bn + bc;
        async_b128(&sB[0][0][br * BP + bc], Pre + h);
        async_b128(&sB[0][1][br * BP + bc], Pim + h);
    }

    for (int i = 0; i < NCH; ++i) {
        const int cur = i & 1;
        if (i + 1 < NCH) {
            // stage next chunk into the other buffer, then wait for current:
            // async completions are in-order, so asynccnt<=4 means the first
            // 4 (current chunk) have landed in LDS.
            const int nxt = cur ^ 1;
            const int kt = (i + 1) * BK;
            const int g = (bm + ar) * NODES + kt + ac;
            async_b128(&sA[nxt][0][ar * AP + ac], Lre + g);
            async_b128(&sA[nxt][1][ar * AP + ac], Lim + g);
            const int h = (kt + br) * NODES + bn + bc;
            async_b128(&sB[nxt][0][br * BP + bc], Pre + h);
            async_b128(&sB[nxt][1][br * BP + bc], Pim + h);
            wait_async_le4();
        } else {
            wait_async_0();
        }
        __syncthreads(); // all waves' chunk-i tiles visible

        const float* __restrict__ aRe = sA[cur][0];
        const float* __restrict__ aIm = sA[cur][1];
        const float* __restrict__ bRe = sB[cur][0];
        const float* __restrict__ bIm = sB[cur][1];

#pragma unroll
        for (int kk = 0; kk < 4; ++kk) {
            const int kb = kk * 4 + half * 2; // this lane-half's K base
            v2f bre, bim, bimn;
            bre.x = bRe[(kb + 0) * BP + wcol + lh];
            bre.y = bRe[(kb + 1) * BP + wcol + lh];
            bim.x = bIm[(kb + 0) * BP + wcol + lh];
            bim.y = bIm[(kb + 1) * BP + wcol + lh];
            bimn.x = -bim.x;  // negate once per k-step (shared by both subtiles)
            bimn.y = -bim.y;
#pragma unroll
            for (int s = 0; s < 2; ++s) {
                const int row = wrow + s * 16 + lh;
                v2f are, aim;
                are.x = aRe[row * AP + kb + 0];
                are.y = aRe[row * AP + kb + 1];
                aim.x = aIm[row * AP + kb + 0];
                aim.y = aIm[row * AP + kb + 1];
                // D_re += Are*Bre + Aim*(-Bim) ; D_im += Are*Bim + Aim*Bre
                acc[2 * s + 0] = wmma4(are, bre,  acc[2 * s + 0]);
                acc[2 * s + 0] = wmma4(aim, bimn, acc[2 * s + 0]);
                acc[2 * s + 1] = wmma4(are, bim,  acc[2 * s + 1]);
                acc[2 * s + 1] = wmma4(aim, bre,  acc[2 * s + 1]);
            }
        }
        __syncthreads(); // all waves done reading buf cur before it is restaged
    }

    // epilogue: D = 2*acc - Q   (C/D layout: row = vgpr + 8*half, col = lh)
#pragma unroll
    for (int s = 0; s < 2; ++s) {
#pragma unroll
        for (int v = 0; v < 8; ++v) {
            const int row = bm + wrow + s * 16 + half * 8 + v;
            const int col = bn + wcol + lh;
            const int g = row * NODES + col;
            Dre[g] = 2.0f * acc[2 * s + 0][v] - Qre[g];
            Dim[g] = 2.0f * acc[2 * s + 1][v] - Qim[g];
        }
    }
}

// ------------------------------------------------------------------ launch
extern "C" void kernel_launch(void* const* d_in, const int* in_sizes, int n_in,
                              void* d_out, int out_size, void* d_ws, size_t ws_size,
                              hipStream_t stream) {
    (void)n_in; (void)out_size; (void)d_ws; (void)ws_size;
    const float* feat  = (const float*)d_in[0];
    const int*   edges = (const int*)d_in[1]; // [2, M] flat: row0=u, row1=v
    const float* w1    = (const float*)d_in[2];
    const float* b1    = (const float*)d_in[3];
    const int M = in_sizes[1] / 2;
    // d_in[4] is K (== 4): out layout fixed as real[5][N][N] then imag[5][N][N]

    float* out = (float*)d_out;
    // Scratch carved out of not-yet-written d_out regions (no ws dependence):
    float* A    = out + 2 * (size_t)NN; // T2-real slot, overwritten by 1st GEMM
    float* p    = out + 3 * (size_t)NN; // T3-real slot (N floats)
    float* dinv = out + 4 * (size_t)NN; // T4-real slot (N floats)

    zero_f32<<<256, 256, 0, stream>>>(A, NN);
    node_potential<<<NODES, IN_DIM, 0, stream>>>(feat, w1, b1, p);
    edge_scatter<<<(M + 255) / 256, 256, 0, stream>>>(edges, edges + M, p, A, M);
    add_eye<<<NODES / 256, 256, 0, stream>>>(A);
    degree_inv_sqrt<<<NODES, 256, 0, stream>>>(A, dinv);
    build_L<<<NN / 256, 256, 0, stream>>>(A, dinv, out);

    dim3 grid(NODES / BM, NODES / BN);
    // T2 = 2*L@T1 - T0
    cheb_cgemm<<<grid, 256, 0, stream>>>(out + 1 * (size_t)NN, out + 6 * (size_t)NN,
                                         out + 1 * (size_t)NN, out + 6 * (size_t)NN,
                                         out + 0 * (size_t)NN, out + 5 * (size_t)NN,
                                         out + 2 * (size_t)NN, out + 7 * (size_t)NN);
    // T3 = 2*L@T2 - T1
    cheb_cgemm<<<grid, 256, 0, stream>>>(out + 1 * (size_t)NN, out + 6 * (size_t)NN,
                                         out + 2 * (size_t)NN, out + 7 * (size_t)NN,
                                         out + 1 * (size_t)NN, out + 6 * (size_t)NN,
                                         out + 3 * (size_t)NN, out + 8 * (size_t)NN);
    // T4 = 2*L@T3 - T2
    cheb_cgemm<<<grid, 256, 0, stream>>>(out + 1 * (size_t)NN, out + 6 * (size_t)NN,
                                         out + 3 * (size_t)NN, out + 8 * (size_t)NN,
                                         out + 2 * (size_t)NN, out + 7 * (size_t)NN,
                                         out + 4 * (size_t)NN, out + 9 * (size_t)NN);
}